// Sparse_Spatial_Attention_4535485464587
// MI455X (gfx1250) — compile-verified
//
#include <hip/hip_runtime.h>
#include <hip/hip_bf16.h>

typedef float v2f __attribute__((ext_vector_type(2)));
typedef float v8f __attribute__((ext_vector_type(8)));
typedef unsigned int v4u __attribute__((ext_vector_type(4)));
typedef int v8i __attribute__((ext_vector_type(8)));
typedef int v4i __attribute__((ext_vector_type(4)));

#define B_  8
#define T_  12
#define N_  512
#define H_  8
#define DH_ 16
#define F_  128
#define KNEI_ 16
#define S_  18          // SAMPLES * log2(N) = 2*9
#define R_  (B_*T_*N_)  // 49152

static __device__ inline v8f wmma_f32_k4(v2f a, v2f b, v8f c) {
    // D = A(16x4) * B(4x16) + C(16x16), full fp32 matrix op (CDNA5)
    return __builtin_amdgcn_wmma_f32_16x16x4_f32(false, a, false, b, (short)0, c,
                                                 false, false);
}

// ---------------------------------------------------------------------------
// eig[n,f] = spa_vec[n,f]*spa_val[f] + tem_vec[n,f]*tem_val[f]
// ---------------------------------------------------------------------------
__global__ __launch_bounds__(256) void eig_kernel(
    const float* __restrict__ sval, const float* __restrict__ svec,
    const float* __restrict__ tval, const float* __restrict__ tvec,
    float* __restrict__ eig) {
    int i = blockIdx.x * 256 + threadIdx.x;   // < N_*F_ = 65536
    int f = i & (F_ - 1);
    eig[i] = svec[i] * sval[f] + tvec[i] * tval[f];
}

// ---------------------------------------------------------------------------
// Generic C[M,128] = act(A[M,128] @ W[128,128] + bias), 128 rows per block,
// 16 rows per wave, full-width 128 cols per wave via 8 accumulator tiles.
// W (64KB) is staged into LDS by the Tensor Data Mover (one 2D D# descriptor,
// tracked by TENSORcnt), issued by wave 0 only.
// MODE 0: plain+bias, 1: A := A + eig[row%N] (QKV projections), 2: ReLU.
// ---------------------------------------------------------------------------
template <int MODE>
__global__ __launch_bounds__(256) void gemm128_kernel(
    const float* __restrict__ A, const float* __restrict__ eig,
    const float* __restrict__ W, const float* __restrict__ bias,
    float* __restrict__ C) {
    __shared__ float Wl[F_ * F_];   // 64 KB
    int tid = threadIdx.x;
    if ((tid >> 5) == 0) {          // wave 0 issues the TDM transfer
        unsigned long long ga = (unsigned long long)(const void*)W;
        unsigned lb = (unsigned)(size_t)(void*)Wl;   // low 32b = LDS offset
        v4u g0;
        g0.x = 1u;                                   // count=1 (user D#)
        g0.y = lb;                                   // lds_addr
        g0.z = (unsigned)ga;                         // global_addr[31:0]
        g0.w = (unsigned)(ga >> 32) | (2u << 30);    // global_addr[56:32]|type=2
        v8i g1;
        g1[0] = (int)(2u << 16);                     // data_size=4B, mask=0
        g1[1] = (int)(128u << 16);                   // tensor_dim0=128 (lo16)
        g1[2] = (int)(128u << 16);                   // td0 hi=0 | tensor_dim1 lo
        g1[3] = (int)(128u << 16);                   // td1 hi=0 | tile_dim0=128
        g1[4] = 128;                                 // tile_dim1=128, tile_dim2=0
        g1[5] = 128;                                 // tensor_dim0_stride=128
        g1[6] = 0;
        g1[7] = 0;
        v4i zz4 = {0, 0, 0, 0};                      // groups 2/3 unused (2D)
        v8i zz8 = {0, 0, 0, 0, 0, 0, 0, 0};
        __builtin_amdgcn_tensor_load_to_lds(g0, g1, zz4, zz4, zz8, 0);
        __builtin_amdgcn_s_wait_tensorcnt((short)0);
    }
    __syncthreads();

    int lane = tid & 31;
    int wv   = tid >> 5;
    int ml   = lane & 15;     // M index within tile / N index of B frag
    int hf   = lane >> 4;     // half-wave selects K pair {0,1} vs {2,3}
    size_t row0 = (size_t)blockIdx.x * 128 + (size_t)wv * 16;
    size_t m    = row0 + ml;

    const float* Ar = A + m * F_;
    const float* Er = (MODE == 1) ? (eig + (m & (N_ - 1)) * F_) : nullptr;
    __builtin_prefetch(Ar, 0, 0);
    __builtin_prefetch(Ar + 64, 0, 0);
    if (MODE == 1) { __builtin_prefetch(Er, 0, 0); __builtin_prefetch(Er + 64, 0, 0); }

    v8f acc[8];
    const v8f vz = {0.f, 0.f, 0.f, 0.f, 0.f, 0.f, 0.f, 0.f};
#pragma unroll
    for (int c = 0; c < 8; ++c) acc[c] = vz;

    for (int i = 0; i < 32; ++i) {          // K loop, 4 per WMMA
        int k = i * 4 + 2 * hf;
        v2f a;
        a.x = Ar[k];
        a.y = Ar[k + 1];
        if (MODE == 1) { a.x += Er[k]; a.y += Er[k + 1]; }
#pragma unroll
        for (int c = 0; c < 8; ++c) {       // 8 col tiles of 16
            v2f b;
            b.x = Wl[k * F_ + c * 16 + ml];        // conflict-free banks
            b.y = Wl[(k + 1) * F_ + c * 16 + ml];
            acc[c] = wmma_f32_k4(a, b, acc[c]);
        }
    }

#pragma unroll
    for (int c = 0; c < 8; ++c) {
        int col  = c * 16 + ml;
        float bv = bias[col];
#pragma unroll
        for (int r = 0; r < 8; ++r) {
            size_t rr = row0 + r + 8 * hf;  // C layout: VGPR r -> M=r / M=r+8
            float v = acc[c][r] + bv;
            if (MODE == 2) v = fmaxf(v, 0.f);
            C[rr * F_ + col] = v;
        }
    }
}

// ---------------------------------------------------------------------------
// M[hb,t,n] = sum_k (Q[hb,t,n,:] . K[hb,t,adj[n,k],:]) * Wp[k] + bp
// ---------------------------------------------------------------------------
__global__ __launch_bounds__(256) void qk_m_kernel(
    const float* __restrict__ Q, const float* __restrict__ Km,
    const int* __restrict__ adj, const float* __restrict__ Wp,
    const float* __restrict__ bp, float* __restrict__ Mout) {
    int idx = blockIdx.x * 256 + threadIdx.x;    // (hb*T + t)*N + n
    int n   = idx & (N_ - 1);
    int bt  = idx >> 9;
    int t   = bt % T_;
    int hb  = bt / T_;
    int h = hb >> 3, b = hb & 7;                 // hb = h*B + b
    size_t rowbase = (size_t)(b * T_ + t) * N_;

    const float* qrow = Q + (rowbase + n) * F_ + h * DH_;
    const float4* q4 = (const float4*)qrow;
    float4 q0 = q4[0], q1 = q4[1], q2 = q4[2], q3 = q4[3];

    float acc = bp[0];
    for (int kk = 0; kk < KNEI_; ++kk) {
        int nbr = adj[n * KNEI_ + kk];
        const float4* k4 = (const float4*)(Km + (rowbase + nbr) * F_ + h * DH_);
        float4 k0 = k4[0], k1 = k4[1], k2 = k4[2], k3 = k4[3];
        float dot = q0.x * k0.x + q0.y * k0.y + q0.z * k0.z + q0.w * k0.w
                  + q1.x * k1.x + q1.y * k1.y + q1.z * k1.z + q1.w * k1.w
                  + q2.x * k2.x + q2.y * k2.y + q2.z * k2.z + q2.w * k2.w
                  + q3.x * k3.x + q3.y * k3.y + q3.z * k3.z + q3.w * k3.w;
        acc = fmaf(dot, Wp[kk], acc);
    }
    Mout[idx] = acc;
}

// ---------------------------------------------------------------------------
// Top-18 of M[hb,t,:] (N=512) — one wave32 per (hb,t), iterative select.
// Order doesn't matter (softmax rows + argmax permute consistently).
// ---------------------------------------------------------------------------
__global__ __launch_bounds__(256) void topk_kernel(
    const float* __restrict__ M, int* __restrict__ Mtop) {
    int lane = threadIdx.x & 31;
    int wv   = blockIdx.x * 8 + (threadIdx.x >> 5);  // < 768
    const float* row = M + (size_t)wv * N_;
    float vals[16];
#pragma unroll
    for (int j = 0; j < 16; ++j) vals[j] = row[j * 32 + lane];

    for (int s = 0; s < S_; ++s) {
        float bv = -__builtin_inff();
        int bi = 0x7fffffff;
#pragma unroll
        for (int j = 0; j < 16; ++j)
            if (vals[j] > bv) { bv = vals[j]; bi = j * 32 + lane; }
#pragma unroll
        for (int off = 16; off > 0; off >>= 1) {
            float ov = __shfl_xor(bv, off, 32);
            int   oi = __shfl_xor(bi, off, 32);
            if (ov > bv || (ov == bv && oi < bi)) { bv = ov; bi = oi; }
        }
        if (lane == 0) Mtop[wv * S_ + s] = bi;
        if ((bi & 31) == lane) vals[bi >> 5] = -__builtin_inff();
    }
}

// ---------------------------------------------------------------------------
// Per-(hb,t) attention: Q_reduce gathered into LDS with per-lane async
// global->LDS loads (ASYNCcnt); scores (WMMA fp32) -> softmax -> argmax_s ->
// attn@V (WMMA fp32, K split across waves) -> gather by cp -> scatter.
// Pad rows use a dummy score row (18) so all LDS stores are branchless.
// ---------------------------------------------------------------------------
__global__ __launch_bounds__(256) void attn_kernel(
    const float* __restrict__ Q, const float* __restrict__ Km,
    const float* __restrict__ Vm, const int* __restrict__ Mtop,
    float* __restrict__ outv) {
    __shared__ __align__(16) float Qr[32 * DH_];   // padded 32x16 Q_reduce (2KB)
    __shared__ float sc[20 * N_];                  // 18 rows + dummy (40KB)
    __shared__ float pb[8 * 256];                  // WMMA partial tiles (8KB)
    __shared__ __align__(16) float ob[S_ * DH_];   // out rows
    __shared__ int cp[N_];

    int blk = blockIdx.x;                          // hb*T + t
    int t = blk % T_, hb = blk / T_;
    int h = hb >> 3, b = hb & 7;
    size_t rowbase = (size_t)(b * T_ + t) * N_;
    int tid = threadIdx.x;
    int lane = tid & 31, wv = tid >> 5;
    int ml = lane & 15, hf = lane >> 4;

    // phase 0: async gather of Q_reduce rows straight into LDS
    // (pad rows 18..31 duplicate row 17); one b128 chunk per lane.
    if (tid < 128) {
        int s = tid >> 2, q = tid & 3;
        int ss = (s < S_) ? s : (S_ - 1);
        int sel = Mtop[blk * S_ + ss];
        const void* ga = (const void*)(Q + (rowbase + sel) * F_ + h * DH_ + q * 4);
        unsigned la = (unsigned)(size_t)(void*)Qr + (unsigned)tid * 16u;
        asm volatile("global_load_async_to_lds_b128 %0, %1, off"
                     :: "v"(la), "v"(ga) : "memory");
        asm volatile("s_wait_asynccnt 0x0" ::: "memory");
    }
    __syncthreads();

    // phase 1: scores = Qr @ K^T / sqrt(Dh); 2 m-tiles x 32 n-tiles, k=16
    for (int p = wv; p < 64; p += 8) {
        int mt = p & 1, nt = p >> 1;
        v8f acc = {0.f, 0.f, 0.f, 0.f, 0.f, 0.f, 0.f, 0.f};
        const float* kbase = Km + (rowbase + nt * 16 + ml) * F_ + h * DH_;
#pragma unroll
        for (int ks = 0; ks < 4; ++ks) {
            int kd = ks * 4 + 2 * hf;
            v2f a;
            a.x = Qr[(mt * 16 + ml) * DH_ + kd];
            a.y = Qr[(mt * 16 + ml) * DH_ + kd + 1];
            v2f bb;                                  // B[kd][n] = K[n][kd]
            bb.x = kbase[kd];
            bb.y = kbase[kd + 1];
            acc = wmma_f32_k4(a, bb, acc);
        }
#pragma unroll
        for (int r = 0; r < 8; ++r) {
            int srow = mt * 16 + r + 8 * hf;
            int sr   = (srow < S_) ? srow : S_;      // pad rows -> dummy row 18
            sc[sr * N_ + nt * 16 + ml] = acc[r] * 0.25f;
        }
    }
    __syncthreads();

    // phase 2: softmax over n for each of 18 rows (wave per row)
    for (int srow = wv; srow < S_; srow += 8) {
        float* rp = sc + srow * N_;
        float v[16];
        float mx = -__builtin_inff();
#pragma unroll
        for (int j = 0; j < 16; ++j) { v[j] = rp[j * 32 + lane]; mx = fmaxf(mx, v[j]); }
#pragma unroll
        for (int off = 16; off > 0; off >>= 1) mx = fmaxf(mx, __shfl_xor(mx, off, 32));
        float sum = 0.f;
#pragma unroll
        for (int j = 0; j < 16; ++j) { v[j] = expf(v[j] - mx); sum += v[j]; }
#pragma unroll
        for (int off = 16; off > 0; off >>= 1) sum += __shfl_xor(sum, off, 32);
        float inv = 1.f / sum;
#pragma unroll
        for (int j = 0; j < 16; ++j) rp[j * 32 + lane] = v[j] * inv;
    }
    __syncthreads();

    // phase 3: cp[n] = argmax_s attn[s,n] (first max, matching jnp.argmax)
    for (int p = tid; p < N_; p += 256) {
        float bv = sc[p];
        int bs = 0;
        for (int s = 1; s < S_; ++s) {
            float v = sc[s * N_ + p];
            if (v > bv) { bv = v; bs = s; }
        }
        cp[p] = bs;
    }

    // phase 4a: out = attn(32x512) @ V(512x16) via WMMA, K split in quarters:
    // wave w -> m-tile (w&1), K-quarter (w>>1); pad A rows clamp to dummy row.
    {
        int mt = wv & 1, kq = wv >> 1;
        int rowA = mt * 16 + ml;
        int rA   = (rowA < S_) ? rowA : S_;          // in-bounds garbage row
        const float* vbase = Vm + rowbase * F_ + h * DH_;
        v8f acc = {0.f, 0.f, 0.f, 0.f, 0.f, 0.f, 0.f, 0.f};
        for (int ks = 0; ks < 32; ++ks) {
            int k = kq * 128 + ks * 4 + 2 * hf;
            v2f a;
            a.x = sc[rA * N_ + k];
            a.y = sc[rA * N_ + k + 1];
            v2f bb;                                  // B[k][d] = V[k][d]
            bb.x = vbase[(size_t)k * F_ + ml];
            bb.y = vbase[(size_t)(k + 1) * F_ + ml];
            acc = wmma_f32_k4(a, bb, acc);
        }
#pragma unroll
        for (int r = 0; r < 8; ++r)
            pb[wv * 256 + (r + 8 * hf) * 16 + ml] = acc[r];
    }
    __syncthreads();

    // phase 4b: reduce the 4 K-quarter partials per m-tile
    for (int p = tid; p < S_ * DH_; p += 256) {
        int s = p >> 4, d = p & 15;
        int mtv = s >> 4, sl = s & 15;
        float v = 0.f;
#pragma unroll
        for (int q = 0; q < 4; ++q) v += pb[(2 * q + mtv) * 256 + sl * 16 + d];
        ob[p] = v;
    }
    __syncthreads();

    // phase 5: value[n,:] = out[cp[n],:] scattered to [b,t,n, h*16+d]
    for (int p = tid; p < N_; p += 256) {
        int s = cp[p];
        float4* dst = (float4*)(outv + (rowbase + p) * F_ + h * DH_);
        const float4* src = (const float4*)(ob + s * DH_);
        dst[0] = src[0]; dst[1] = src[1]; dst[2] = src[2]; dst[3] = src[3];
    }
}

// ---------------------------------------------------------------------------
// LayerNorm over F=128, one wave per row.
// AFFINE: *g+b.  ADD: input = X + X2 (residual).
// ---------------------------------------------------------------------------
template <bool AFFINE, bool ADD>
__global__ __launch_bounds__(256) void ln_kernel(
    const float* __restrict__ X, const float* __restrict__ X2,
    const float* __restrict__ g, const float* __restrict__ bb,
    float* __restrict__ out) {
    int lane = threadIdx.x & 31;
    size_t row = (size_t)blockIdx.x * 8 + (threadIdx.x >> 5);
    const float* xp = X + row * F_;
    float x[4];
#pragma unroll
    for (int j = 0; j < 4; ++j) {
        int c = j * 32 + lane;
        x[j] = xp[c];
        if (ADD) x[j] += X2[row * F_ + c];
    }
    float s = x[0] + x[1] + x[2] + x[3];
#pragma unroll
    for (int off = 16; off > 0; off >>= 1) s += __shfl_xor(s, off, 32);
    float mu = s * (1.f / F_);
    float vs = 0.f;
#pragma unroll
    for (int j = 0; j < 4; ++j) { float d = x[j] - mu; vs += d * d; }
#pragma unroll
    for (int off = 16; off > 0; off >>= 1) vs += __shfl_xor(vs, off, 32);
    float rstd = rsqrtf(vs * (1.f / F_) + 1e-5f);
#pragma unroll
    for (int j = 0; j < 4; ++j) {
        int c = j * 32 + lane;
        float v = (x[j] - mu) * rstd;
        if (AFFINE) v = v * g[c] + bb[c];
        out[row * F_ + c] = v;
    }
}

// ---------------------------------------------------------------------------
extern "C" void kernel_launch(void* const* d_in, const int* in_sizes, int n_in,
                              void* d_out, int out_size, void* d_ws, size_t ws_size,
                              hipStream_t stream) {
    (void)in_sizes; (void)n_in; (void)out_size; (void)ws_size;
    const float* x    = (const float*)d_in[0];
    const float* sval = (const float*)d_in[1];
    const float* svec = (const float*)d_in[2];
    const float* tval = (const float*)d_in[3];
    const float* tvec = (const float*)d_in[4];
    const int*   ladj = (const int*)d_in[5];
    const float* Wq = (const float*)d_in[6];  const float* bq = (const float*)d_in[7];
    const float* Wk = (const float*)d_in[8];  const float* bk = (const float*)d_in[9];
    const float* Wv = (const float*)d_in[10]; const float* bv = (const float*)d_in[11];
    const float* Wo = (const float*)d_in[12]; const float* bo = (const float*)d_in[13];
    const float* lng = (const float*)d_in[14]; const float* lnb = (const float*)d_in[15];
    const float* W1 = (const float*)d_in[16]; const float* b1 = (const float*)d_in[17];
    const float* W2 = (const float*)d_in[18]; const float* b2 = (const float*)d_in[19];
    const float* Wp = (const float*)d_in[20]; const float* bp = (const float*)d_in[21];
    float* out = (float*)d_out;

    const size_t RF = (size_t)R_ * F_;          // 6,291,456
    float* eig     = (float*)d_ws;              // N*F
    float* Qb      = eig + (size_t)N_ * F_;
    float* Kb      = Qb + RF;
    float* Vb      = Kb + RF;
    float* Mb      = Vb + RF;                   // HB*T*N = 393216
    int*   Mtop    = (int*)(Mb + (size_t)64 * T_ * N_);
    float* attnval = (float*)(Mtop + (size_t)64 * T_ * S_ + 32 /*pad*/);
    float* vout    = attnval + RF;              // also post-LN "value"
    float* hbuf    = Qb;                        // reuse Q for FFN hidden
    float* ybuf    = Kb;                        // reuse K for FFN output

    eig_kernel<<<(N_ * F_) / 256, 256, 0, stream>>>(sval, svec, tval, tvec, eig);
    gemm128_kernel<1><<<R_ / 128, 256, 0, stream>>>(x, eig, Wq, bq, Qb);
    gemm128_kernel<1><<<R_ / 128, 256, 0, stream>>>(x, eig, Wk, bk, Kb);
    gemm128_kernel<1><<<R_ / 128, 256, 0, stream>>>(x, eig, Wv, bv, Vb);
    qk_m_kernel<<<(64 * T_ * N_) / 256, 256, 0, stream>>>(Qb, Kb, ladj, Wp, bp, Mb);
    topk_kernel<<<(64 * T_) / 8, 256, 0, stream>>>(Mb, Mtop);
    attn_kernel<<<64 * T_, 256, 0, stream>>>(Qb, Kb, Vb, Mtop, attnval);
    gemm128_kernel<0><<<R_ / 128, 256, 0, stream>>>(attnval, nullptr, Wo, bo, vout);
    ln_kernel<true, false><<<R_ / 8, 256, 0, stream>>>(vout, nullptr, lng, lnb, vout);
    gemm128_kernel<2><<<R_ / 128, 256, 0, stream>>>(vout, nullptr, W1, b1, hbuf);
    gemm128_kernel<0><<<R_ / 128, 256, 0, stream>>>(hbuf, nullptr, W2, b2, ybuf);
    ln_kernel<false, true><<<R_ / 8, 256, 0, stream>>>(ybuf, vout, nullptr, nullptr, out);
}